// OnlineLabelSmoothing_3710851743799
// MI455X (gfx1250) — compile-verified
//
#include <hip/hip_runtime.h>

// Online Label Smoothing forward + stats for MI455X (gfx1250, wave32).
// d_in[0]: x [32768,1000] f32, d_in[1]: target [32768] int, d_in[2]: matrix [1000,1000] f32
// d_out: [ loss(1) | grad_acc(1000*1000) | count_acc(1000) ] f32

#define NUM_CLASSES 1000
#define BATCH       32768
#define K_ITERS     (NUM_CLASSES / 4)   // 250 steps of K=4 fp32 WMMA

typedef __attribute__((ext_vector_type(2))) float v2f;
typedef __attribute__((ext_vector_type(8))) float v8f;

__global__ void ols_zero_kernel(float* __restrict__ out, int n) {
    int i = blockIdx.x * blockDim.x + threadIdx.x;
    int stride = gridDim.x * blockDim.x;
    for (; i < n; i += stride) out[i] = 0.0f;
}

__global__ __launch_bounds__(256) void ols_main_kernel(
    const float* __restrict__ x,
    const int*   __restrict__ target,
    const float* __restrict__ matrix,
    float*       __restrict__ out)
{
    float* loss_out  = out;
    float* grad_acc  = out + 1;
    float* count_acc = out + 1 + (size_t)NUM_CLASSES * NUM_CLASSES;

    const int lane = threadIdx.x & 31;
    const int wave = threadIdx.x >> 5;
    const int tile = blockIdx.x * (blockDim.x >> 5) + wave;  // 2048 tiles of 16 rows
    const int r0   = tile * 16;
    const int m    = lane & 15;            // batch row within tile owned by this lane
    const int koff = (lane >> 4) << 1;     // 0 for lanes 0-15, 2 for lanes 16-31

    const int row = r0 + m;
    const int tgt = target[row];

    const float* xrow = x      + (size_t)row * NUM_CLASSES + koff;
    const float* mrow = matrix + (size_t)tgt * NUM_CLASSES + koff;

    // ---- Pass 1: WMMA-accumulated gathered dot; row max/argmax; softlabel row sum
    v8f  acc  = {};
    float mx   = -3.0e38f;
    int   amax = 0;
    float ssum = 0.0f;

    for (int i = 0; i < K_ITERS; ++i) {
        const int k = 4 * i;
        v2f a = *(const v2f*)(xrow + k);   // A: x[r0+m, k+koff .. +1]
        v2f b = *(const v2f*)(mrow + k);   // B: matrix[t_m, k+koff .. +1]

        const int c0 = k + koff;
        bool g0 = a.x > mx;  mx = g0 ? a.x : mx;  amax = g0 ? c0     : amax;
        bool g1 = a.y > mx;  mx = g1 ? a.y : mx;  amax = g1 ? c0 + 1 : amax;
        ssum += b.x + b.y;

        // C[m][n] += sum_k x[r0+m,k] * matrix[t_n,k]; diagonal = per-row dot
        acc = __builtin_amdgcn_wmma_f32_16x16x4_f32(
                  false, a, false, b, (short)0, acc, false, false);
    }

    // Fold lane <-> lane^16 partials (each pair covers one full row)
    {
        float mx2  = __shfl_xor(mx, 16);
        int   am2  = __shfl_xor(amax, 16);
        bool take = (mx2 > mx) || (mx2 == mx && am2 < amax);  // first-occurrence ties
        mx   = take ? mx2 : mx;
        amax = take ? am2 : amax;
        ssum += __shfl_xor(ssum, 16);
    }

    // ---- Pass 2: sum of exp(x - max); x tile is L2-hot from pass 1
    float se = 0.0f;
    for (int i = 0; i < K_ITERS; ++i) {
        v2f a = *(const v2f*)(xrow + 4 * i);
        se += __expf(a.x - mx) + __expf(a.y - mx);
    }
    se += __shfl_xor(se, 16);

    // ---- Extract diagonal of C (dot per row) and route to row owners.
    // C layout: C[M][N]: M<8 -> VGPR M, lane N; M>=8 -> VGPR M-8, lane 16+N.
    // Diagonal lives in lanes 0..7 (rows 0..7, element=lane) and lanes 24..31 (rows 8..15).
    float diag = 0.0f;
    const int sel = lane & 7;
#pragma unroll
    for (int i = 0; i < 8; ++i) { if (sel == i) diag = acc[i]; }
    const int src = (m < 8) ? m : (m + 16);
    float dot = __shfl(diag, src);

    // ---- Loss: sum_c softlabel*(-logprobs) = (max+log(se))*sum(softlabel) - dot
    float lse    = mx + __logf(se);
    float loss_m = (lane < 16) ? (lse * ssum - dot) : 0.0f;
#pragma unroll
    for (int off = 16; off > 0; off >>= 1) loss_m += __shfl_xor(loss_m, off);
    if (lane == 0) atomicAdd(loss_out, loss_m * (1.0f / (float)BATCH));

    // ---- Rare path: rows where argmax == target (~B/1000 rows).
    // Whole wave cooperates per correct row: scatter softmax probs + count.
    bool correct = (lane < 16) && (amax == tgt);
    unsigned int mask = (unsigned int)__ballot(correct);
    float inv_se = 1.0f / se;

    while (mask) {
        int mm = __ffs(mask) - 1;    // lane index == row-in-tile (restricted to <16)
        mask &= mask - 1;
        int   crow = r0 + mm;
        int   ct   = __shfl(tgt, mm);
        float cmx  = __shfl(mx, mm);
        float cinv = __shfl(inv_se, mm);
        const float* xr = x + (size_t)crow * NUM_CLASSES;
        float* grow = grad_acc + (size_t)ct * NUM_CLASSES;
        for (int cc = lane; cc < NUM_CLASSES; cc += 32) {
            float p = __expf(xr[cc] - cmx) * cinv;
            atomicAdd(&grow[cc], p);
        }
        if (lane == 0) atomicAdd(&count_acc[ct], 1.0f);
    }
}

extern "C" void kernel_launch(void* const* d_in, const int* in_sizes, int n_in,
                              void* d_out, int out_size, void* d_ws, size_t ws_size,
                              hipStream_t stream) {
    const float* x      = (const float*)d_in[0];
    const int*   target = (const int*)  d_in[1];
    const float* matrix = (const float*)d_in[2];
    float*       out    = (float*)d_out;

    // Zero loss + grad_acc + count_acc (harness poisons d_out).
    ols_zero_kernel<<<512, 256, 0, stream>>>(out, out_size);

    // 2048 tiles of 16 rows; 8 waves (256 threads) per block -> 256 blocks.
    const int tiles = BATCH / 16;
    ols_main_kernel<<<tiles / 8, 256, 0, stream>>>(x, target, matrix, out);
}